// myGATConv_6648609374674
// MI455X (gfx1250) — compile-verified
//
#include <hip/hip_runtime.h>
#include <math.h>

#define NN   50000
#define EE   800000
#define INF_ 256
#define HH   8
#define DHD  32
#define HD   256          // HH*DHD
#define NEC  8
#define LRA  0.2f

typedef __attribute__((ext_vector_type(2))) float v2f;
typedef __attribute__((ext_vector_type(8))) float v8f;

// ---------------------------------------------------------------------------
// Kernel 1: fused dual GEMM via V_WMMA_F32_16X16X4_F32 with 1x8 register
// blocking: each wave computes a 16x128 output strip (8 accumulators), so the
// A fragment loaded per k-step feeds 8 WMMAs. feat is read only 4x total
// (~200 MB HBM) instead of 32x (~1.6 GB).
//   col groups 0,1 (cols   0..255): h   = feat @ W_fc  + b_fc   -> hbuf
//   col groups 2,3 (cols 256..511): res = feat @ W_res + b_res  -> d_out
// 50000 % 16 == 0. 12500 waves total.
// ---------------------------------------------------------------------------
__global__ __launch_bounds__(256) void gat_gemm_dual(
    const float* __restrict__ feat,
    const float* __restrict__ Wfc,  const float* __restrict__ bfc,
    const float* __restrict__ Wres, const float* __restrict__ bres,
    float* __restrict__ hbuf, float* __restrict__ outbuf)
{
    const int wv   = threadIdx.x >> 5;
    const int lane = threadIdx.x & 31;
    const int half = lane >> 4;        // 0: lanes 0-15, 1: lanes 16-31
    const int lr   = lane & 15;

    const int w = blockIdx.x * 8 + wv;          // wave id, 0 .. 12499
    if (w >= 12500) return;                     // wave-uniform: EXEC stays full
    const int mt = w >> 2;                      // m tile, 0 .. 3124
    const int cg = w & 3;                       // column group of 128
    const int m0 = mt * 16;

    const float* W    = (cg < 2) ? Wfc  : Wres;
    const float* bia  = (cg < 2) ? bfc  : bres;
    float*       dstp = (cg < 2) ? hbuf : outbuf;
    const int colbase = (cg & 1) * 128;

    v8f acc[8] = {};
    const float* arow = feat + (size_t)(m0 + lr) * INF_;   // A row for this lane
    const float* brow = W + colbase + lr;

    for (int k = 0; k < INF_; k += 4) {
        // A 16x4 layout: VGPR0 = K{0|2}, VGPR1 = K{1|3} (half selects pair)
        v2f a;
        a.x = arow[k + 2 * half];
        a.y = arow[k + 2 * half + 1];
        const float* b0 = brow + (size_t)(k + 2 * half) * HD;
        #pragma unroll
        for (int t = 0; t < 8; ++t) {
            // B 4x16 layout (symmetric): lane%16 = N, VGPR+half = K
            v2f b;
            b.x = b0[t * 16];
            b.y = b0[t * 16 + HD];
            acc[t] = __builtin_amdgcn_wmma_f32_16x16x4_f32(
                         false, a, false, b, (short)0, acc[t], false, false);
        }
    }

    #pragma unroll
    for (int t = 0; t < 8; ++t) {
        const int col = colbase + t * 16 + lr;
        const float bv = bia[col];
        #pragma unroll
        for (int v = 0; v < 8; ++v) {
            const int row = m0 + v + 8 * half;   // C/D layout: VGPR v -> M=v / M=8+v
            dstp[(size_t)row * HD + col] = acc[t][v] + bv;
        }
    }
}

// ---------------------------------------------------------------------------
// Order-preserving float <-> uint encoding for integer atomicMax on floats.
// ---------------------------------------------------------------------------
__device__ __forceinline__ unsigned enc_f(float f) {
    unsigned u = __float_as_uint(f);
    return (u & 0x80000000u) ? ~u : (u | 0x80000000u);
}
__device__ __forceinline__ float dec_f(unsigned k) {
    unsigned u = (k & 0x80000000u) ? (k ^ 0x80000000u) : ~k;
    return __uint_as_float(u);
}

// ---------------------------------------------------------------------------
// Kernel 2: per-node attention scores s_src/s_dst; init mx (enc) and den.
// ---------------------------------------------------------------------------
__global__ void gat_node_scores(const float* __restrict__ hbuf,
                                const float* __restrict__ attn,
                                float* __restrict__ s_src, float* __restrict__ s_dst,
                                unsigned* __restrict__ mx_enc, float* __restrict__ den)
{
    const int idx = blockIdx.x * blockDim.x + threadIdx.x;
    if (idx >= NN * HH) return;
    const int n  = idx >> 3;
    const int hd = idx & 7;
    const float* hp = hbuf + (size_t)n * HD + hd * DHD;
    const float* as = attn + hd * (3 * DHD);        // attn[0, hd, 0:32]
    const float* ad = as + DHD;                     // attn[0, hd, 32:64]
    float ss = 0.f, sd = 0.f;
    #pragma unroll
    for (int d = 0; d < DHD; ++d) { const float x = hp[d]; ss += x * as[d]; sd += x * ad[d]; }
    s_src[idx]  = ss;
    s_dst[idx]  = sd;
    mx_enc[idx] = 0u;      // encodes minimal ordering key
    den[idx]    = 0.f;
}

// ---------------------------------------------------------------------------
// Kernel 3: per-edge-type attention score table (8 types x 8 heads).
// ---------------------------------------------------------------------------
__global__ void gat_etype_scores(const float* __restrict__ edge_emb,
                                 const float* __restrict__ attn,
                                 float* __restrict__ s_e)
{
    const int t = threadIdx.x;
    if (t >= NEC * HH) return;
    const int et = t >> 3, hd = t & 7;
    const float* ep = edge_emb + et * HD + hd * DHD;
    const float* ae = attn + hd * (3 * DHD) + 2 * DHD;  // attn[0, hd, 64:96]
    float s = 0.f;
    #pragma unroll
    for (int d = 0; d < DHD; ++d) s += ep[d] * ae[d];
    s_e[t] = s;
}

// ---------------------------------------------------------------------------
// Kernel 4: edge scores (leaky-relu) + segment max via encoded atomicMax.
// ---------------------------------------------------------------------------
__global__ void gat_edge_scores(const int* __restrict__ src, const int* __restrict__ dst,
                                const int* __restrict__ et,
                                const float* __restrict__ s_src, const float* __restrict__ s_dst,
                                const float* __restrict__ s_e,
                                float* __restrict__ score, unsigned* __restrict__ mx_enc)
{
    const int idx = blockIdx.x * blockDim.x + threadIdx.x;
    if (idx >= EE * HH) return;
    const int e  = idx >> 3;
    const int hd = idx & 7;
    const int s = src[e], d = dst[e], t = et[e];
    float sc = s_src[s * 8 + hd] + s_dst[d * 8 + hd] + s_e[t * 8 + hd];
    sc = (sc > 0.f) ? sc : LRA * sc;
    score[idx] = sc;
    atomicMax(&mx_enc[d * 8 + hd], enc_f(sc));
}

// ---------------------------------------------------------------------------
// Kernel 5: ex = exp(score - mx[dst]) (in place); den = segment_sum(ex).
// ---------------------------------------------------------------------------
__global__ void gat_edge_exp(const int* __restrict__ dst,
                             const unsigned* __restrict__ mx_enc,
                             float* __restrict__ score, float* __restrict__ den)
{
    const int idx = blockIdx.x * blockDim.x + threadIdx.x;
    if (idx >= EE * HH) return;
    const int e  = idx >> 3;
    const int hd = idx & 7;
    const int d  = dst[e];
    const float ex = expf(score[idx] - dec_f(mx_enc[d * 8 + hd]));
    score[idx] = ex;
    atomicAdd(&den[d * 8 + hd], ex);
}

// ---------------------------------------------------------------------------
// Kernel 6: message aggregation. 64 lanes x float4 per edge, 4 edges/block.
// out already holds the residual (from gat_gemm_dual).
// ---------------------------------------------------------------------------
__global__ __launch_bounds__(256) void gat_edge_msg(
    const int* __restrict__ src, const int* __restrict__ dst,
    const float* __restrict__ hbuf, const float* __restrict__ ex,
    const float* __restrict__ den, float* __restrict__ out)
{
    const int e = blockIdx.x * 4 + (threadIdx.x >> 6);
    if (e >= EE) return;
    const int t    = threadIdx.x & 63;
    const int base = t * 4;            // 4 consecutive floats, same head (32 | 4)
    const int hd   = base >> 5;
    const int s = src[e], d = dst[e];
    const float w = ex[e * 8 + hd] / den[d * 8 + hd];
    const float4 hv = *(const float4*)(hbuf + (size_t)s * HD + base);
    float* op = out + (size_t)d * HD + base;
    atomicAdd(op + 0, hv.x * w);
    atomicAdd(op + 1, hv.y * w);
    atomicAdd(op + 2, hv.z * w);
    atomicAdd(op + 3, hv.w * w);
}

// ---------------------------------------------------------------------------
// Kernel 7: ELU in place.
// ---------------------------------------------------------------------------
__global__ void gat_elu(float* __restrict__ out)
{
    const int i = blockIdx.x * blockDim.x + threadIdx.x;
    if (i >= NN * HD) return;
    const float x = out[i];
    out[i] = (x > 0.f) ? x : expm1f(x);
}

// ---------------------------------------------------------------------------
extern "C" void kernel_launch(void* const* d_in, const int* in_sizes, int n_in,
                              void* d_out, int out_size, void* d_ws, size_t ws_size,
                              hipStream_t stream)
{
    const float* feat = (const float*)d_in[0];
    const int*   src  = (const int*)  d_in[1];
    const int*   dst  = (const int*)  d_in[2];
    const int*   et   = (const int*)  d_in[3];
    const float* Wfc  = (const float*)d_in[4];
    const float* bfc  = (const float*)d_in[5];
    const float* eemb = (const float*)d_in[6];
    const float* attn = (const float*)d_in[7];
    const float* Wres = (const float*)d_in[8];
    const float* bres = (const float*)d_in[9];
    float* out = (float*)d_out;

    char* ws = (char*)d_ws;
    float*    hbuf  = (float*)ws;     ws += sizeof(float) * (size_t)NN * HD;      // 51.2 MB
    float*    s_src = (float*)ws;     ws += sizeof(float) * (size_t)NN * HH;      // 1.6 MB
    float*    s_dst = (float*)ws;     ws += sizeof(float) * (size_t)NN * HH;
    unsigned* mx    = (unsigned*)ws;  ws += sizeof(unsigned) * (size_t)NN * HH;
    float*    den   = (float*)ws;     ws += sizeof(float) * (size_t)NN * HH;
    float*    s_e   = (float*)ws;     ws += sizeof(float) * NEC * HH;
    float*    score = (float*)ws;     ws += sizeof(float) * (size_t)EE * HH;      // 25.6 MB

    // 1. dual GEMM: h -> hbuf, residual -> out  (12500 waves, 8 per block)
    gat_gemm_dual<<<(12500 + 7) / 8, 256, 0, stream>>>(feat, Wfc, bfc, Wres, bres, hbuf, out);

    // 2. per-node scores + init of mx/den
    gat_node_scores<<<(NN * HH + 255) / 256, 256, 0, stream>>>(hbuf, attn, s_src, s_dst, mx, den);

    // 3. per-edge-type score table
    gat_etype_scores<<<1, 64, 0, stream>>>(eemb, attn, s_e);

    // 4. edge scores + segment max
    gat_edge_scores<<<(EE * HH + 255) / 256, 256, 0, stream>>>(src, dst, et, s_src, s_dst, s_e, score, mx);

    // 5. exp + segment sum
    gat_edge_exp<<<(EE * HH + 255) / 256, 256, 0, stream>>>(dst, mx, score, den);

    // 6. weighted message scatter-add
    gat_edge_msg<<<EE / 4, 256, 0, stream>>>(src, dst, hbuf, score, den, out);

    // 7. ELU in place
    gat_elu<<<((size_t)NN * HD + 255) / 256, 256, 0, stream>>>(out);
}